// BasisOrbitalBackflow_12429635355200
// MI455X (gfx1250) — compile-verified
//
#include <hip/hip_runtime.h>

typedef float v2f __attribute__((ext_vector_type(2)));
typedef float v8f __attribute__((ext_vector_type(8)));

#define NPART 14
#define SDIM  3
#define NB    20
#define KTOT  (NB * NB)   // 400
#define NPAD  16
#define WAVES_PER_BLOCK 4

// NS multi-index table: itertools.product(range(4),repeat=3), sum<=3, stable-sorted by sum
__device__ __constant__ int c_ns0[NB] = {0, 0,0,1, 0,0,0,1,1,2, 0,0,0,0,1,1,1,2,2,3};
__device__ __constant__ int c_ns1[NB] = {0, 0,1,0, 0,1,2,0,1,0, 0,1,2,3,0,1,2,0,1,0};
__device__ __constant__ int c_ns2[NB] = {0, 1,0,0, 2,1,0,1,0,0, 3,2,1,0,2,1,0,1,0,0};

__device__ __forceinline__ float hermite_sel(float x, int n) {
    // physicists' Hermite H_n(x), n in [0,3]
    float h1 = 2.0f * x;
    float h2 = 4.0f * x * x - 2.0f;
    float h3 = (8.0f * x * x - 12.0f) * x;
    return (n == 0) ? 1.0f : (n == 1) ? h1 : (n == 2) ? h2 : h3;
}

__global__ __launch_bounds__(128) void
bf_wmma_kernel(const float* __restrict__ ri,
               const float* __restrict__ sigma,
               const float* __restrict__ coeff,
               float* __restrict__ out,
               int nbatch) {
    // coeff staged as interleaved K-pairs: element (K,o) at (K>>1)*32 + o*2 + (K&1)
    // -> the two B-frag elements of one WMMA step are one aligned b64 apart.
    __shared__ float coeffLds[KTOT * NPAD];                 // 400x16, cols 14/15 zero
    __shared__ float basisLds[WAVES_PER_BLOCK][NPAD * NB];  // per-wave 16 x 20
    __shared__ float aLds[WAVES_PER_BLOCK][NPAD * NB];      // per-wave 16 x 20

    const int tid  = threadIdx.x;
    const int wave = tid >> 5;
    const int lane = tid & 31;
    const int braw = blockIdx.x * WAVES_PER_BLOCK + wave;
    const bool valid = (braw < nbatch);
    const int b = valid ? braw : 0;   // clamp: duplicate work, skip store

    // ---- stage coeff (400x14) into LDS in K-pair-interleaved layout ----
    for (int i = tid; i < KTOT * NPAD; i += 128) {
        int kp = i >> 5;            // K pair
        int rem = i & 31;
        int o  = rem >> 1;
        int lo = rem & 1;
        int K  = kp * 2 + lo;
        coeffLds[i] = (o < NPART) ? coeff[K * NPART + o] : 0.0f;
    }

    // ---- per-wave: basis[n][mu] = Hx*Hy*Hz * exp(-0.5*sigma_mu^2*r^2) ----
    float* myB = basisLds[wave];
    float* myA = aLds[wave];
    for (int idx = lane; idx < NPAD * NB; idx += 32) {
        int n = idx / NB, mu = idx % NB;
        float v = 0.0f;
        if (n < NPART) {
            const float* r = ri + (size_t)(b * NPART + n) * SDIM;
            float x = r[0], y = r[1], z = r[2];
            float s = sigma[mu];
            float r2 = x * x + y * y + z * z;
            v = hermite_sel(x, c_ns0[mu]) * hermite_sel(y, c_ns1[mu]) *
                hermite_sel(z, c_ns2[mu]) * __expf(-0.5f * s * s * r2);
        }
        myB[idx] = v;
    }
    __syncthreads();

    // ---- A[n][p] = (S[p] - basis[n][p]) / (NPART-1), rows >= NPART zeroed ----
    if (lane < NB) {
        int p = lane;
        float S = 0.0f;
        #pragma unroll
        for (int n = 0; n < NPART; ++n) S += myB[n * NB + p];
        const float inv = 1.0f / (float)(NPART - 1);
        #pragma unroll
        for (int n = 0; n < NPAD; ++n)
            myA[n * NB + p] = (n < NPART) ? (S - myB[n * NB + p]) * inv : 0.0f;
    }
    __syncthreads();

    // ---- WMMA: D(16x16) = pooled(16x400) @ coeffPad(400x16), K steps of 4 ----
    // Lane holds row M (A) / col N (B) = lane&15; K = 4*ks + 2*half + {0,1}.
    // p = (4ks)/20 is half-independent and compile-time; q = (4ks)%20 + 2*half.
    const int half = lane >> 4;
    const int l15  = lane & 15;

    // Preload per-lane operand rows into registers (constant-indexed under unroll)
    float rA[NB];
    #pragma unroll
    for (int p = 0; p < NB; ++p) rA[p] = myA[l15 * NB + p];

    float bq[10];                       // basis values at q = 4j + 2*half + {0,1}
    #pragma unroll
    for (int j = 0; j < 5; ++j) {
        bq[2 * j]     = myB[l15 * NB + 4 * j + 2 * half];
        bq[2 * j + 1] = myB[l15 * NB + 4 * j + 2 * half + 1];
    }

    const float* cBase = coeffLds + (half * 32 + l15 * 2);  // 8B-aligned per lane

    v8f acc = {};
    #pragma unroll
    for (int ks = 0; ks < KTOT / 4; ++ks) {
        const int p = (4 * ks) / NB;    // compile-time constant
        const int j = ks % 5;           // compile-time constant
        v2f afrag, bfrag;
        afrag.x = rA[p] * bq[2 * j];
        afrag.y = rA[p] * bq[2 * j + 1];
        bfrag = *(const v2f*)(cBase + ks * 64);  // ds_load_b64, imm offset ks*256
        acc = __builtin_amdgcn_wmma_f32_16x16x4_f32(false, afrag, false, bfrag,
                                                    (short)0, acc, false, false);
    }

    // ---- store D: VGPR r -> (M = r + 8*half, N = l15); keep M,N < 14 ----
    if (valid && l15 < NPART) {
        #pragma unroll
        for (int r = 0; r < 8; ++r) {
            int M = r + 8 * half;
            if (M < NPART)
                out[((size_t)b * NPART + M) * NPART + l15] = acc[r];
        }
    }
}

extern "C" void kernel_launch(void* const* d_in, const int* in_sizes, int n_in,
                              void* d_out, int out_size, void* d_ws, size_t ws_size,
                              hipStream_t stream) {
    // inputs (setup_inputs order): ri, rij_dist (unused), sigma, coeff
    const float* ri    = (const float*)d_in[0];
    const float* sigma = (const float*)d_in[2];
    const float* coeff = (const float*)d_in[3];
    float* out = (float*)d_out;

    int nbatch = in_sizes[0] / (NPART * SDIM);  // 2048
    int blocks = (nbatch + WAVES_PER_BLOCK - 1) / WAVES_PER_BLOCK;
    bf_wmma_kernel<<<blocks, 128, 0, stream>>>(ri, sigma, coeff, out, nbatch);
}